// RoiAlign_9423158247477
// MI455X (gfx1250) — compile-verified
//
#include <hip/hip_runtime.h>

// ROI-Align (FPN, 7x7, C=256) for gfx1250.
// Bandwidth-bound gather: x-lerp fused into the gather (VALU, data-dependent
// addresses), y-lerp expressed as a shared-weight 16x16x4 f32 WMMA per box.
// Output uses non-temporal stores so the ~179MB feature pyramid stays
// resident in the 192MB L2 while the 100MB write-once output streams past it.

typedef float v2f __attribute__((ext_vector_type(2)));
typedef float v8f __attribute__((ext_vector_type(8)));

#define CROP   7
#define C_CH   256

__global__ __launch_bounds__(128) void roi_align_wmma_kernel(
    const float* __restrict__ boxes,
    const float* __restrict__ f0, const float* __restrict__ f1,
    const float* __restrict__ f2, const float* __restrict__ f3,
    const float* __restrict__ f4,
    const int*   __restrict__ ishape,
    float*       __restrict__ out,
    int B, int N)
{
    // Per-wave staging tile: H[r=16 (14 used + 2 zero pad)][j=7][cc=16]
    __shared__ float Hs[4][16][CROP][16];

    const int lane = threadIdx.x & 31;
    const int wave = threadIdx.x >> 5;
    const int bn   = blockIdx.x >> 2;   // box index in [0, B*N)
    const int cblk = blockIdx.x & 3;    // 64-channel block
    const int b    = bn / N;
    const int c0   = cblk * 64 + wave * 16;

    // ---- box + FPN level (uniform per block) ----
    const float bx1 = boxes[bn * 4 + 0];
    const float by1 = boxes[bn * 4 + 1];
    const float bx2 = boxes[bn * 4 + 2];
    const float by2 = boxes[bn * 4 + 3];

    const float area  = (float)ishape[0] * (float)ishape[1];
    const float ratio = sqrtf((by2 - by1) * (bx2 - bx1)) * sqrtf(area) / 56.0f;
    int lvl = (int)rintf(log2f(ratio));          // rintf = round-half-even (jnp.round)
    lvl = lvl < 0 ? 0 : (lvl > 4 ? 4 : lvl);

    const float* F; int S;
    switch (lvl) {
        case 0:  F = f0; S = 256; break;
        case 1:  F = f1; S = 128; break;
        case 2:  F = f2; S = 64;  break;
        case 3:  F = f3; S = 32;  break;
        default: F = f4; S = 16;  break;
    }
    F += (size_t)b * (size_t)S * (size_t)S * C_CH;

    const float Hf = (float)(S - 1);
    const float Wf = Hf;
    const float sy = (by2 - by1) * Hf / 6.0f;
    const float sx = (bx2 - bx1) * Wf / 6.0f;
    const float ybase = by1 * Hf;
    const float xbase = bx1 * Wf;

    const int cc = lane & 15;   // channel within 16-group
    const int jj = lane >> 4;   // half-wave selector

    // ---- zero K-padding rows 14,15 of the staging tile ----
    {
        const int r = 14 + jj;
        #pragma unroll
        for (int j = 0; j < CROP; ++j) Hs[wave][r][j][cc] = 0.0f;
    }

    // ---- gather + fused x-lerp: H[r][j][cc] for r = 2i + {0,1} ----
    for (int r = 0; r < 14; ++r) {
        const int   i   = r >> 1;
        const float ysv = ybase + (float)i * sy;
        float yr = floorf(ysv) + (float)(r & 1);
        yr = fminf(fmaxf(yr, 0.0f), Hf);
        const float* Frow = F + (size_t)(int)yr * (size_t)S * C_CH;

        #pragma unroll
        for (int jb = 0; jb < 4; ++jb) {
            const int j = jb * 2 + jj;
            if (j < CROP) {
                const float xsv = xbase + (float)j * sx;
                const float x0f = floorf(xsv);
                const float lx  = xsv - x0f;
                const int xi0 = (int)fminf(fmaxf(x0f, 0.0f), Wf);
                const int xi1 = (int)fminf(fmaxf(x0f + 1.0f, 0.0f), Wf);
                const float va = Frow[(size_t)xi0 * C_CH + c0 + cc];
                const float vb = Frow[(size_t)xi1 * C_CH + c0 + cc];
                float h = fmaf(lx, vb - va, va);
                if (!(xsv >= 0.0f && xsv <= Wf)) h = 0.0f;   // x out-of-range mask
                Hs[wave][r][j][cc] = h;
            }
        }
    }
    __syncthreads();

    // ---- A fragments: Wy[m][r] = (r==2m)?(1-ly) : (r==2m+1)?ly : 0 ----
    // A 16x4 f32 layout: lanes 0-15 = M rows; elem0 = K{base}, elem1 = K{base+1},
    // lanes 16-31 shifted by +2 in K.
    const int   m    = lane & 15;
    const float ysm  = ybase + (float)m * sy;
    const float lym  = ysm - floorf(ysm);
    const bool  yok  = (m < CROP) && (ysm >= 0.0f) && (ysm <= Hf);  // y mask
    const float w0   = yok ? (1.0f - lym) : 0.0f;
    const float w1   = yok ? lym : 0.0f;
    const int   kOff = jj * 2;

    v2f A[4];
    #pragma unroll
    for (int q = 0; q < 4; ++q) {
        const int rA = q * 4 + kOff;
        float a0 = (rA     == 2 * m) ? w0 : ((rA     == 2 * m + 1) ? w1 : 0.0f);
        float a1 = (rA + 1 == 2 * m) ? w0 : ((rA + 1 == 2 * m + 1) ? w1 : 0.0f);
        A[q].x = a0; A[q].y = a1;
    }

    // ---- y-lerp via V_WMMA_F32_16X16X4_F32, then NT store ----
    const size_t outBase = (size_t)bn * (CROP * CROP) * C_CH + c0;
    for (int j = 0; j < CROP; ++j) {
        v8f acc = {};
        #pragma unroll
        for (int q = 0; q < 4; ++q) {
            const int kb = q * 4 + kOff;
            v2f Bv;
            Bv.x = Hs[wave][kb + 0][j][cc];
            Bv.y = Hs[wave][kb + 1][j][cc];
            acc = __builtin_amdgcn_wmma_f32_16x16x4_f32(
                false, A[q], false, Bv, (short)0, acc, false, false);
        }
        // D rows v+8*half; only rows 0..6 (half==0, lanes 0-15) are real output.
        if (lane < 16) {
            #pragma unroll
            for (int v = 0; v < CROP; ++v) {
                __builtin_nontemporal_store(
                    acc[v], out + outBase + (size_t)(v * CROP + j) * C_CH + cc);
            }
        }
    }
}

extern "C" void kernel_launch(void* const* d_in, const int* in_sizes, int n_in,
                              void* d_out, int out_size, void* d_ws, size_t ws_size,
                              hipStream_t stream) {
    (void)n_in; (void)out_size; (void)d_ws; (void)ws_size;

    const float* boxes = (const float*)d_in[0];
    const float* f0    = (const float*)d_in[1];
    const float* f1    = (const float*)d_in[2];
    const float* f2    = (const float*)d_in[3];
    const float* f3    = (const float*)d_in[4];
    const float* f4    = (const float*)d_in[5];
    const int*   ish   = (const int*)d_in[6];
    float*       out   = (float*)d_out;

    const int BN = in_sizes[0] / 4;                       // B*N
    int B = in_sizes[1] / (256 * 256 * 256);              // feat0 = B*256*256*256
    if (B <= 0) B = 1;
    const int N = BN / B;

    dim3 grid((unsigned)(BN * 4));   // box * 4 channel-blocks of 64
    dim3 block(128);                 // 4 waves (wave32); wave -> 16 channels
    hipLaunchKernelGGL(roi_align_wmma_kernel, grid, block, 0, stream,
                       boxes, f0, f1, f2, f3, f4, ish, out, B, N);
}